// NNC1_62405874811858
// MI455X (gfx1250) — compile-verified
//
#include <hip/hip_runtime.h>
#include <hip/hip_bf16.h>

// ---------------------------------------------------------------------------
// NNConv GNN forward for MI455X (gfx1250, wave32).
//
// msg[e] = x[src[e]] @ reshape(h[e]@w2 + b2, [ci,co]) fused into one dense
// GEMM per layer: A[e, kk*ci+i] = h[e,kk]*x[src[e],i] synthesized from LDS,
// B = w2 reshaped [32*ci, co] (+ one ci-block for b2), via
// V_WMMA_F32_16X16X4_F32 (fp32 in/accum = reference precision).
// 64-edge M-tile per block: each staged B pair feeds 4 WMMAs (4 accumulators)
// so B LDS traffic and L2 re-reads are amortized 4x. All staging guards are
// branch-free (clamped address + select), so the hot loop is ds+valu+wmma.
// ---------------------------------------------------------------------------

typedef __attribute__((ext_vector_type(2))) float v2f;
typedef __attribute__((ext_vector_type(8))) float v8f;

#define NN_NODES 15000
#define NN_EDGES 30000
#define NN_GRAPHS 8
#define NN_EMB 40
#define NN_CLS 2
#define NN_HID 32

// ---- monotone float <-> uint key for atomic max pooling --------------------
__device__ __forceinline__ unsigned fkey(float f) {
    unsigned b = __float_as_uint(f);
    return (b & 0x80000000u) ? ~b : (b | 0x80000000u);
}
__device__ __forceinline__ float fdecode(unsigned u) {
    unsigned b = (u & 0x80000000u) ? (u & 0x7FFFFFFFu) : ~u;
    return __uint_as_float(b);
}

// ---- out[n,c] = sum_i x[n,i]*root[i,c] + bias[c]  (initializes scatter dst)
__global__ void node_root_kernel(const float* __restrict__ xin,
                                 const float* __restrict__ root,
                                 const float* __restrict__ bias,
                                 float* __restrict__ xout, int ci, int co) {
    int t = blockIdx.x * blockDim.x + threadIdx.x;
    if (t >= NN_NODES * co) return;
    int nd = t / co, c = t - nd * co;
    float s = bias[c];
    for (int i = 0; i < ci; ++i) s += xin[nd * ci + i] * root[i * co + c];
    xout[nd * co + c] = s;
}

// ---- h[e,k] = relu(edge_attr[e]*w1[k] + b1[k]) -----------------------------
__global__ void edge_h_kernel(const float* __restrict__ ea,
                              const float* __restrict__ w1,
                              const float* __restrict__ b1,
                              float* __restrict__ h) {
    int t = blockIdx.x * blockDim.x + threadIdx.x;
    if (t >= NN_EDGES * NN_HID) return;
    int e = t >> 5, k = t & 31;
    h[t] = fmaxf(ea[e] * w1[k] + b1[k], 0.0f);
}

// ---- fused edge GEMM + scatter-add (the heavy kernel) ----------------------
// Block = 64-edge tile (4 M-subtiles), CO_T waves; wave w owns cols [16w,16w+16).
// Per kk in [0,32]: wave stages a zero-padded (CI_PAD x 16) B tile in private
// LDS (branch-free), then per 4-wide K chunk reuses the B fragment across the
// 4 M-subtiles -> 4 back-to-back WMMAs per B load.
// f32 WMMA fragments (ISA 7.12.2): A lane<16 rows K{0,1}, lane>=16 K{2,3};
// B symmetric over columns; D: VGPR r -> row r / r+8, col = lane&15.
template <int CI, int CO>
__global__ __launch_bounds__(32 * ((CO + 15) / 16))
void edge_gemm_wmma_kernel(const float* __restrict__ h,
                           const float* __restrict__ xin,
                           const int* __restrict__ src,
                           const int* __restrict__ dst,
                           const float* __restrict__ w2,
                           const float* __restrict__ b2,
                           float* __restrict__ xout) {
    constexpr int CI_PAD = (CI + 3) & ~3;      // 4, 32, 64
    constexpr int XSTR   = CI_PAD + 3;         // odd stride: conflict-free
    constexpr int CO_T   = (CO + 15) / 16;     // waves per block
    constexpr int MT     = 4;                  // M-subtiles per block
    constexpr int TM     = 16 * MT;            // 64 edges per block

    __shared__ float hs[TM * 33];              // 64 edges x 32 hidden
    __shared__ float xs[TM * XSTR];            // 64 gathered x[src] rows
    __shared__ float bs[CO_T][CI_PAD * 17];    // per-wave B tile (stride 17)
    __shared__ int sidx[TM];
    __shared__ int didx[TM];

    const int lane = threadIdx.x;
    const int wave = threadIdx.y;
    const int tid  = wave * 32 + lane;
    constexpr int NT = 32 * CO_T;
    const int e0 = blockIdx.x * TM;

    for (int idx = tid; idx < TM; idx += NT) {
        int e = e0 + idx; e = (e < NN_EDGES) ? e : (NN_EDGES - 1);  // clamp
        sidx[idx] = src[e];
        didx[idx] = dst[e];
    }
    for (int idx = tid; idx < TM * NN_HID; idx += NT) {
        int r = idx >> 5, k = idx & 31;
        int e = e0 + r; e = (e < NN_EDGES) ? e : (NN_EDGES - 1);
        hs[r * 33 + k] = h[e * NN_HID + k];
    }
    __syncthreads();                           // sidx visible to all waves
    for (int idx = tid; idx < TM * CI_PAD; idx += NT) {
        int r = idx / CI_PAD, i = idx - r * CI_PAD;
        int isafe = (CI == CI_PAD) ? i : ((i < CI) ? i : CI - 1);
        float v = xin[sidx[r] * CI + isafe];   // always-legal load
        xs[r * XSTR + i] = (i < CI) ? v : 0.0f;
    }
    __syncthreads();

    const int m = lane & 15;                   // A row / B,D column index
    const int khalf = (lane >> 4) << 1;        // 0 or 2
    const int cbase = wave * 16;
    const int c = cbase + m;                   // this lane's output column
    float* __restrict__ bsw = bs[wave];

    v8f acc[MT] = {};
    for (int kk = 0; kk <= NN_HID; ++kk) {     // kk == 32 -> b2 bias block
        const float* __restrict__ brow =
            (kk < NN_HID) ? (w2 + (size_t)kk * (CI * CO)) : b2;
        float hv[MT];
#pragma unroll
        for (int t = 0; t < MT; ++t)
            hv[t] = (kk < NN_HID) ? hs[(t * 16 + m) * 33 + kk] : 1.0f;

        // stage zero-padded B tile (branch-free: clamped address + select);
        // in-order DS within the wave -> no barrier between staging and use
#pragma unroll
        for (int idx0 = 0; idx0 < CI_PAD * 16; idx0 += 32) {
            const int idx = idx0 + lane;
            const int i = idx >> 4, ncol = idx & 15;
            const int cc = cbase + ncol;
            const int isafe = (CI == CI_PAD) ? i : ((i < CI) ? i : CI - 1);
            const int csafe = (CO % 16 == 0) ? cc : ((cc < CO) ? cc : CO - 1);
            float v = brow[isafe * CO + csafe];
            if (CI != CI_PAD || CO % 16 != 0)
                v = (i < CI && cc < CO) ? v : 0.0f;
            bsw[i * 17 + ncol] = v;
        }
        // guard-free WMMA chain: B fragment reused across 4 M-subtiles
#pragma unroll
        for (int i0 = 0; i0 < CI_PAD; i0 += 4) {
            const int ia = i0 + khalf;
            v2f b;
            b.x = bsw[ia * 17 + m];
            b.y = bsw[(ia + 1) * 17 + m];
#pragma unroll
            for (int t = 0; t < MT; ++t) {
                v2f a;
                a.x = hv[t] * xs[(t * 16 + m) * XSTR + ia];
                a.y = hv[t] * xs[(t * 16 + m) * XSTR + ia + 1];
                acc[t] = __builtin_amdgcn_wmma_f32_16x16x4_f32(
                    false, a, false, b, (short)0, acc[t], false, false);
            }
        }
    }

    // scatter-add msg tiles into xout[dst[e], c]  (segment_sum over dst)
    const int rbase = (lane >> 4) * 8;
#pragma unroll
    for (int t = 0; t < MT; ++t) {
        if (e0 + t * 16 < NN_EDGES) {          // whole subtile valid (E%16==0)
#pragma unroll
            for (int r = 0; r < 8; ++r) {
                if (c < CO)
                    atomicAdd(&xout[(size_t)didx[t * 16 + rbase + r] * CO + c],
                              acc[t][r]);
            }
        }
    }
}

__global__ void relu_kernel(float* __restrict__ p, int n) {
    int t = blockIdx.x * blockDim.x + threadIdx.x;
    if (t < n) p[t] = fmaxf(p[t], 0.0f);
}

__global__ void pool_init_kernel(unsigned* __restrict__ keys) {
    int t = blockIdx.x * blockDim.x + threadIdx.x;
    if (t < NN_GRAPHS * NN_EMB) keys[t] = 0u;   // below every real float key
}

__global__ void pool_max_kernel(const float* __restrict__ x,
                                const int* __restrict__ batch,
                                unsigned* __restrict__ keys) {
    int t = blockIdx.x * blockDim.x + threadIdx.x;
    if (t >= NN_NODES * NN_EMB) return;
    int nd = t / NN_EMB, k = t - nd * NN_EMB;
    atomicMax(&keys[batch[nd] * NN_EMB + k], fkey(x[t]));
}

__global__ void head_kernel(const unsigned* __restrict__ keys,
                            const float* __restrict__ fcw,
                            const float* __restrict__ fcb,
                            float* __restrict__ out) {
    int t = blockIdx.x * blockDim.x + threadIdx.x;
    if (t >= NN_GRAPHS * NN_CLS) return;
    int g = t / NN_CLS, c = t - g * NN_CLS;
    float s = fcb[c];
    for (int k = 0; k < NN_EMB; ++k) {
        unsigned u = keys[g * NN_EMB + k];
        float v = (u == 0u) ? 0.0f : fmaxf(fdecode(u), 0.0f);  // relu(seg_max)
        s += v * fcw[k * NN_CLS + c];
    }
    out[t] = s;
}

// ---------------------------------------------------------------------------
extern "C" void kernel_launch(void* const* d_in, const int* in_sizes, int n_in,
                              void* d_out, int out_size, void* d_ws, size_t ws_size,
                              hipStream_t stream) {
    (void)in_sizes; (void)n_in; (void)out_size; (void)ws_size;

    const float* x0    = (const float*)d_in[0];   // [15000,3]
    const int*   eidx  = (const int*)d_in[1];     // [2,30000]
    const float* ea    = (const float*)d_in[2];   // [30000,1]
    const int*   batch = (const int*)d_in[3];     // [15000]
    const int* src = eidx;
    const int* dst = eidx + NN_EDGES;

    // params pytree flattening: conv0..conv4 each {b1,b2,bias,root,w1,w2}
    // (sorted keys), then fc_b, fc_w.
    const float* B1[5]; const float* B2[5]; const float* BIAS[5];
    const float* ROOT[5]; const float* W1[5]; const float* W2[5];
    for (int l = 0; l < 5; ++l) {
        int base = 4 + 6 * l;
        B1[l]   = (const float*)d_in[base + 0];
        B2[l]   = (const float*)d_in[base + 1];
        BIAS[l] = (const float*)d_in[base + 2];
        ROOT[l] = (const float*)d_in[base + 3];
        W1[l]   = (const float*)d_in[base + 4];
        W2[l]   = (const float*)d_in[base + 5];
    }
    const float* fcb = (const float*)d_in[34];
    const float* fcw = (const float*)d_in[35];

    // workspace: two node-feature ping-pong buffers (N x 64 max), h, pool keys
    float* xA = (float*)d_ws;
    float* xB = xA + (size_t)NN_NODES * 64;
    float* hb = xB + (size_t)NN_NODES * 64;
    unsigned* keys = (unsigned*)(hb + (size_t)NN_EDGES * NN_HID);

    const int CI[5] = {3, 32, 32, 64, 64};
    const int CO[5] = {32, 32, 64, 64, 40};
    const int NTILE = (NN_EDGES + 63) / 64;   // 469 (last tile partial)

    const float* xin = x0;
    float* bufs[2] = {xA, xB};
    for (int l = 0; l < 5; ++l) {
        const int ci = CI[l], co = CO[l];
        float* xout = bufs[l & 1];

        // 1) initialize with root term: xout = xin @ root + bias
        {
            int nt = NN_NODES * co;
            node_root_kernel<<<(nt + 255) / 256, 256, 0, stream>>>(
                xin, ROOT[l], BIAS[l], xout, ci, co);
        }
        // 2) edge hidden: h = relu(ea * w1 + b1)
        {
            int nt = NN_EDGES * NN_HID;
            edge_h_kernel<<<(nt + 255) / 256, 256, 0, stream>>>(
                ea, W1[l], B1[l], hb);
        }
        // 3) fused (h ⊗ x[src]) @ [w2;b2] with f32 WMMA + atomic scatter-add
        switch (l) {
        case 0:
            edge_gemm_wmma_kernel<3, 32><<<NTILE, dim3(32, 2), 0, stream>>>(
                hb, xin, src, dst, W2[l], B2[l], xout);
            break;
        case 1:
            edge_gemm_wmma_kernel<32, 32><<<NTILE, dim3(32, 2), 0, stream>>>(
                hb, xin, src, dst, W2[l], B2[l], xout);
            break;
        case 2:
            edge_gemm_wmma_kernel<32, 64><<<NTILE, dim3(32, 4), 0, stream>>>(
                hb, xin, src, dst, W2[l], B2[l], xout);
            break;
        case 3:
            edge_gemm_wmma_kernel<64, 64><<<NTILE, dim3(32, 4), 0, stream>>>(
                hb, xin, src, dst, W2[l], B2[l], xout);
            break;
        default:
            edge_gemm_wmma_kernel<64, 40><<<NTILE, dim3(32, 3), 0, stream>>>(
                hb, xin, src, dst, W2[l], B2[l], xout);
            break;
        }
        // 4) relu between layers (not after conv4)
        if (l < 4) {
            int nt = NN_NODES * co;
            relu_kernel<<<(nt + 255) / 256, 256, 0, stream>>>(xout, nt);
        }
        xin = xout;
    }

    // global max pool (segment_max over batch) via monotone-key atomicMax
    pool_init_kernel<<<1, 512, 0, stream>>>(keys);
    {
        int nt = NN_NODES * NN_EMB;
        pool_max_kernel<<<(nt + 255) / 256, 256, 0, stream>>>(xin, batch, keys);
    }
    // relu(emb) @ fc_w + fc_b
    head_kernel<<<1, 32, 0, stream>>>(keys, fcw, fcb, (float*)d_out);
}